// ChannelAttention_28243704939042
// MI455X (gfx1250) — compile-verified
//
#include <hip/hip_runtime.h>
#include <hip/hip_bf16.h>
#include <math.h>

#define BATCH 32
#define DIM   64
#define WLEN  8192
#define HEADS 8
#define HD    8

#define TILE  128
#define HALO  8
#define EXT   144   // HALO + TILE + 8 : 9 chunks of 16, divisible by 4

typedef float v2f __attribute__((ext_vector_type(2)));
typedef float v8f __attribute__((ext_vector_type(8)));

// workspace layout (floats)
#define XSEQ_OFF  0
#define ALPHA_OFF (BATCH * WLEN)
#define BETA_OFF  (ALPHA_OFF + BATCH * DIM)
#define PQ_OFF    (BETA_OFF + BATCH * DIM)
#define G_OFF     (PQ_OFF + BATCH * 2)
#define H_OFF     (G_OFF + BATCH * DIM)

__device__ __forceinline__ v8f wmma_f32(v2f a, v2f b, v8f c) {
  // D = A(16x4 f32) * B(4x16 f32) + C(16x16 f32)
  return __builtin_amdgcn_wmma_f32_16x16x4_f32(false, a, false, b, (short)0, c,
                                               false, false);
}

__device__ __forceinline__ int chm_of(int h, int m) {
  // gram row m in [0,16): rows 0..7 = q channels, 8..15 = k channels
  return (m < 8) ? (h * HD + m) : (DIM + h * HD + (m - 8));
}

// Guarded staging (boundary tiles): zero-pad out-of-range positions.
__device__ __forceinline__ void stage_tile_guard(const float* __restrict__ xb,
                                                 float* s_dst, int t0, int tid) {
  for (int i = tid; i < DIM * EXT; i += 256) {
    int c = i / EXT, e = i - c * EXT;
    int w = t0 - HALO + e;
    s_dst[i] = (w >= 0 && w < WLEN) ? xb[c * WLEN + w] : 0.0f;
  }
}

// Interior staging: async DMA global -> LDS (B128), ASYNCcnt-tracked.
// 2304 16-byte chunks = 9 per thread; EXT%4==0 so chunks never straddle rows.
__device__ __forceinline__ void stage_tile_async(const float* __restrict__ xb,
                                                 float* s_dst, int t0, int tid) {
#pragma unroll
  for (int j = 0; j < 9; ++j) {
    int idx = j * 256 + tid;
    int c = idx / (EXT / 4);
    int e4 = (idx - c * (EXT / 4)) << 2;
    const float* g = xb + c * WLEN + (t0 - HALO) + e4;
    unsigned lds = (unsigned)(uintptr_t)(&s_dst[c * EXT + e4]);
    asm volatile("global_load_async_to_lds_b128 %0, %1, off"
                 :
                 : "v"(lds), "v"(g)
                 : "memory");
  }
}

// -------------------------------------------------------------------------
// K1: xseq[b,w] = sum_c sp_w[c] * dwconv(qkv_v)[c,w] + sp_b
// -------------------------------------------------------------------------
__global__ __launch_bounds__(256) void k1_xseq(
    const float* __restrict__ x, const float* __restrict__ qkv_w,
    const float* __restrict__ qkv_b, const float* __restrict__ dw_w,
    const float* __restrict__ dw_b, const float* __restrict__ sp_w,
    const float* __restrict__ sp_b, float* __restrict__ ws) {
  __shared__ float s_x[DIM * EXT];
  __shared__ float s_v[DIM * EXT];
  __shared__ float s_red[256];
  const int tid = threadIdx.x;
  const int lane = tid & 31;
  const int wv = tid >> 5;
  const int b = blockIdx.y;
  const int t0 = blockIdx.x * TILE;
  const float* xb = x + (size_t)b * DIM * WLEN;

  if (blockIdx.x >= 1 && blockIdx.x <= 62) {
    stage_tile_async(xb, s_x, t0, tid);
    asm volatile("s_wait_asynccnt 0" ::: "memory");
  } else {
    stage_tile_guard(xb, s_x, t0, tid);
  }
  __syncthreads();

  const int mrow = lane & 15;
  const int khalf = (lane >> 4) << 1;  // lanes 0-15 -> K 0,1 ; 16-31 -> K 2,3
  for (int tile = wv; tile < 36; tile += 8) {   // 4 M-chunks x 9 N-chunks
    int mc = tile / 9, nc = tile - mc * 9;
    int ochan = 128 + mc * 16 + mrow;           // v third of qkv_w
    int n = nc * 16 + mrow;
    v8f acc = {};
#pragma unroll
    for (int ks = 0; ks < 16; ++ks) {
      int kc = ks * 4 + khalf;
      v2f a, bb;
      a.x = qkv_w[ochan * DIM + kc];
      a.y = qkv_w[ochan * DIM + kc + 1];
      bb.x = s_x[kc * EXT + n];
      bb.y = s_x[(kc + 1) * EXT + n];
      acc = wmma_f32(a, bb, acc);
    }
    int w = t0 - HALO + n;
    bool inw = (w >= 0 && w < WLEN);
    int mbase = (lane < 16) ? 0 : 8;
#pragma unroll
    for (int r = 0; r < 8; ++r) {
      int cv = mc * 16 + mbase + r;  // 0..63 (v channel)
      float val = acc[r] + qkv_b[128 + cv];
      s_v[cv * EXT + n] = inw ? val : 0.0f;  // conv zero-pads biased signal
    }
  }
  __syncthreads();

  {
    int wi = tid & 127;
    int half = tid >> 7;
    int e = HALO + wi;
    float p = 0.0f;
    for (int c = half * 32; c < half * 32 + 32; ++c) {
      int ch = 128 + c;
      float w0 = dw_w[ch * 9 + 3], w1 = dw_w[ch * 9 + 4], w2 = dw_w[ch * 9 + 5];
      float vd = w0 * s_v[c * EXT + e - 1] + w1 * s_v[c * EXT + e] +
                 w2 * s_v[c * EXT + e + 1] + dw_b[ch];
      p += sp_w[c] * vd;
    }
    s_red[tid] = p;
  }
  __syncthreads();
  if (tid < 128)
    ws[XSEQ_OFF + b * WLEN + t0 + tid] = s_red[tid] + s_red[tid + 128] + sp_b[0];
}

// -------------------------------------------------------------------------
// K2: per (b,h): Gram of [q;k] rows via WMMA (A==B fragments), softmax,
//     fold mp_w/mp_b -> alpha,beta.  Double-buffered async x staging.
// -------------------------------------------------------------------------
__global__ __launch_bounds__(256) void k2_attn(
    const float* __restrict__ x, const float* __restrict__ qkv_w,
    const float* __restrict__ qkv_b, const float* __restrict__ dw_w,
    const float* __restrict__ dw_b, const float* __restrict__ temperature,
    const float* __restrict__ mp_w, const float* __restrict__ mp_b,
    float* __restrict__ ws) {
  __shared__ float s_x[2][DIM * EXT];
  __shared__ float s_r[16 * EXT];
  __shared__ float s_red[8 * 256];
  __shared__ float s_S[256];
  const int tid = threadIdx.x;
  const int lane = tid & 31;
  const int wv = tid >> 5;
  const int h = blockIdx.x;
  const int b = blockIdx.y;
  const float* xb = x + (size_t)b * DIM * WLEN;

  const int mrow = lane & 15;
  const int khalf = (lane >> 4) << 1;

  // Preload A fragments: 16 qkv_w rows for this head's q,k channels.
  v2f aA[16];
  float dt0, dt1, dt2, dtb;
  {
    int ch = chm_of(h, mrow);
#pragma unroll
    for (int ks = 0; ks < 16; ++ks) {
      int kc = ks * 4 + khalf;
      aA[ks].x = qkv_w[ch * DIM + kc];
      aA[ks].y = qkv_w[ch * DIM + kc + 1];
    }
    dt0 = dw_w[ch * 9 + 3]; dt1 = dw_w[ch * 9 + 4]; dt2 = dw_w[ch * 9 + 5];
    dtb = dw_b[ch];
  }

  // Prologue: tile 0 is a boundary tile -> guarded direct staging.
  stage_tile_guard(xb, s_x[0], 0, tid);

  v8f S = {};
  for (int t = 0; t < 64; ++t) {
    int t0 = t * TILE;
    // Issue next tile's staging (async DMA for interior tiles 1..62).
    if (t + 1 < 64) {
      int t0n = (t + 1) * TILE;
      if (t + 1 <= 62)
        stage_tile_async(xb, s_x[(t + 1) & 1], t0n, tid);
      else
        stage_tile_guard(xb, s_x[(t + 1) & 1], t0n, tid);
    }
    // Wait for current tile's async loads (in-order completion: leaving at
    // most the 9 just-issued ops outstanding drains everything older).
    if (t <= 61)
      asm volatile("s_wait_asynccnt 9" ::: "memory");
    else
      asm volatile("s_wait_asynccnt 0" ::: "memory");
    __syncthreads();

    const float* sx = s_x[t & 1];
    // GEMM: 16 channels x EXT columns into s_r
    for (int nc = wv; nc < 9; nc += 8) {
      int n = nc * 16 + mrow;
      v8f acc = {};
#pragma unroll
      for (int ks = 0; ks < 16; ++ks) {
        int kc = ks * 4 + khalf;
        v2f bb;
        bb.x = sx[kc * EXT + n];
        bb.y = sx[(kc + 1) * EXT + n];
        acc = wmma_f32(aA[ks], bb, acc);
      }
      int w = t0 - HALO + n;
      bool inw = (w >= 0 && w < WLEN);
      int mbase = (lane < 16) ? 0 : 8;
#pragma unroll
      for (int r = 0; r < 8; ++r) {
        int m = mbase + r;
        float val = acc[r] + qkv_b[chm_of(h, m)];
        s_r[m * EXT + n] = inw ? val : 0.0f;
      }
    }
    __syncthreads();

    // Gram accumulate: this wave's 16 interior columns, K=4 per WMMA.
    {
      int ebase = HALO + wv * 16;
      const float* rp = &s_r[mrow * EXT];
#pragma unroll
      for (int s4 = 0; s4 < 4; ++s4) {
        int e0 = ebase + s4 * 4 + khalf;
        float d0 = dt0 * rp[e0 - 1] + dt1 * rp[e0] + dt2 * rp[e0 + 1] + dtb;
        float d1 = dt0 * rp[e0] + dt1 * rp[e0 + 1] + dt2 * rp[e0 + 2] + dtb;
        v2f f; f.x = d0; f.y = d1;
        S = wmma_f32(f, f, S);  // S = R * R^T : A and B fragments identical
      }
    }
  }

  // Reduce the 8 per-wave partial Grams.
#pragma unroll
  for (int r = 0; r < 8; ++r) s_red[wv * 256 + r * 32 + lane] = S[r];
  __syncthreads();
  {
    float v = 0.0f;
    for (int w8 = 0; w8 < 8; ++w8) v += s_red[w8 * 256 + tid];
    int r = tid >> 5, l = tid & 31;
    int m = r + ((l & 16) ? 8 : 0);
    int n = l & 15;
    s_S[m * 16 + n] = v;
  }
  __syncthreads();

  if (tid < 8) {
    int c = tid;
    float qn = fmaxf(sqrtf(s_S[c * 16 + c]), 1e-12f);
    float temp = temperature[h];
    float logit[8];
    float mx = -1e30f;
#pragma unroll
    for (int d = 0; d < 8; ++d) {
      float kn = fmaxf(sqrtf(s_S[(8 + d) * 16 + (8 + d)]), 1e-12f);
      logit[d] = s_S[c * 16 + 8 + d] / (qn * kn) * temp;
      mx = fmaxf(mx, logit[d]);
    }
    float sum = 0.0f;
#pragma unroll
    for (int d = 0; d < 8; ++d) { logit[d] = expf(logit[d] - mx); sum += logit[d]; }
    float inv = 1.0f / sum;
    float alpha = 0.0f, beta = 0.0f;
#pragma unroll
    for (int d = 0; d < 8; ++d) {
      float a = logit[d] * inv;
      alpha += a * mp_w[h * HD + d];
      beta += a * mp_b[h * HD + d];
    }
    ws[ALPHA_OFF + b * DIM + h * HD + c] = alpha;
    ws[BETA_OFF + b * DIM + h * HD + c] = beta;
  }
}

// -------------------------------------------------------------------------
// K3: three MTL branches (conv k=2/4/6 -> LN(W) -> ELU -> mean) -> P,Q
// -------------------------------------------------------------------------
__global__ __launch_bounds__(256) void k3_mtl(
    const float* __restrict__ up_w, const float* __restrict__ up_b,
    const float* __restrict__ c2_w, const float* __restrict__ c2_b,
    const float* __restrict__ ln2_w, const float* __restrict__ ln2_b,
    const float* __restrict__ c4_w, const float* __restrict__ c4_b,
    const float* __restrict__ ln4_w, const float* __restrict__ ln4_b,
    const float* __restrict__ c6_w, const float* __restrict__ c6_b,
    const float* __restrict__ ln6_w, const float* __restrict__ ln6_b,
    float* __restrict__ ws) {
  __shared__ float s_xs[WLEN];
  __shared__ float s_y[WLEN];
  __shared__ float s_ra[256];
  __shared__ float s_rb[256];
  const int tid = threadIdx.x;
  const int b = blockIdx.x;
  for (int i = tid; i < WLEN; i += 256) s_xs[i] = ws[XSEQ_OFF + b * WLEN + i];
  __syncthreads();

  const float* cws[3] = {c2_w, c4_w, c6_w};
  const float* cbs[3] = {c2_b, c4_b, c6_b};
  const float* lws[3] = {ln2_w, ln4_w, ln6_w};
  const float* lbs[3] = {ln2_b, ln4_b, ln6_b};
  const int klen[3] = {2, 4, 6};
  const int los[3] = {0, 1, 2};

  float P = 0.0f, Q = 0.0f;
  for (int j = 0; j < 3; ++j) {
    const float* cw = cws[j];
    float cb = cbs[j][0];
    float uw = up_w[j], ub = up_b[j];
    int k = klen[j], lo = los[j];
    float sum = 0.0f, sq = 0.0f;
    for (int w = tid; w < WLEN; w += 256) {
      float y = cb;
      for (int tt = 0; tt < k; ++tt) {
        int src = w + tt - lo;
        float u = (src >= 0 && src < WLEN) ? (uw * s_xs[src] + ub) : 0.0f;
        y += cw[tt] * u;
      }
      s_y[w] = y;
      sum += y;
      sq += y * y;
    }
    s_ra[tid] = sum; s_rb[tid] = sq;
    __syncthreads();
    for (int off = 128; off > 0; off >>= 1) {
      if (tid < off) { s_ra[tid] += s_ra[tid + off]; s_rb[tid] += s_rb[tid + off]; }
      __syncthreads();
    }
    float mu = s_ra[0] / (float)WLEN;
    float var = s_rb[0] / (float)WLEN - mu * mu;
    float rstd = rsqrtf(var + 1e-5f);
    __syncthreads();

    const float* lw = lws[j];
    const float* lb = lbs[j];
    float es = 0.0f;
    for (int w = tid; w < WLEN; w += 256) {
      float z = (s_y[w] - mu) * rstd * lw[w] + lb[w];
      es += (z > 0.0f) ? z : expm1f(z);
    }
    s_ra[tid] = es;
    __syncthreads();
    for (int off = 128; off > 0; off >>= 1) {
      if (tid < off) s_ra[tid] += s_ra[tid + off];
      __syncthreads();
    }
    float sj = s_ra[0] / (float)WLEN;
    P += sj * uw;
    Q += sj * ub;
    __syncthreads();
  }
  if (tid == 0) {
    ws[PQ_OFF + b * 2 + 0] = P;
    ws[PQ_OFF + b * 2 + 1] = Q;
  }
}

// -------------------------------------------------------------------------
// K3b: fold proj + alpha/beta + P,Q -> G,H  (rank-1 final form)
// -------------------------------------------------------------------------
__global__ __launch_bounds__(64) void k3b_combine(
    const float* __restrict__ proj_w, const float* __restrict__ proj_b,
    float* __restrict__ ws) {
  const int b = blockIdx.x;
  const int o = threadIdx.x;
  float A = 0.0f, C = proj_b[o];
  for (int ch = 0; ch < DIM; ++ch) {
    float pw = proj_w[o * DIM + ch];
    A += pw * ws[ALPHA_OFF + b * DIM + ch];
    C += pw * ws[BETA_OFF + b * DIM + ch];
  }
  float P = ws[PQ_OFF + b * 2], Q = ws[PQ_OFF + b * 2 + 1];
  ws[G_OFF + b * DIM + o] = A * P;
  ws[H_OFF + b * DIM + o] = A * Q + C;
}

// -------------------------------------------------------------------------
// K4: out[b,o,w] = G[b,o]*xseq[b,w] + H[b,o]   (streaming float4 stores)
// -------------------------------------------------------------------------
__global__ __launch_bounds__(256) void k4_out(const float* __restrict__ ws,
                                              float* __restrict__ out) {
  __shared__ float s_g[DIM], s_h[DIM];
  const int tid = threadIdx.x;
  const int b = blockIdx.y;
  const int chunk = blockIdx.x;  // 8 chunks of 1024
  if (tid < DIM) {
    s_g[tid] = ws[G_OFF + b * DIM + tid];
    s_h[tid] = ws[H_OFF + b * DIM + tid];
  }
  __syncthreads();
  int w0 = chunk * 1024 + tid * 4;
  float4 xs = *reinterpret_cast<const float4*>(&ws[XSEQ_OFF + b * WLEN + w0]);
  float* ob = out + (size_t)b * DIM * WLEN;
#pragma unroll 4
  for (int o = 0; o < DIM; ++o) {
    float g = s_g[o], h = s_h[o];
    float4 r;
    r.x = g * xs.x + h; r.y = g * xs.y + h;
    r.z = g * xs.z + h; r.w = g * xs.w + h;
    *reinterpret_cast<float4*>(&ob[o * WLEN + w0]) = r;
  }
}

extern "C" void kernel_launch(void* const* d_in, const int* in_sizes, int n_in,
                              void* d_out, int out_size, void* d_ws,
                              size_t ws_size, hipStream_t stream) {
  (void)in_sizes; (void)n_in; (void)out_size; (void)ws_size;
  const float* x = (const float*)d_in[0];
  const float* qkv_w = (const float*)d_in[1];
  const float* qkv_b = (const float*)d_in[2];
  const float* dw_w = (const float*)d_in[3];
  const float* dw_b = (const float*)d_in[4];
  const float* proj_w = (const float*)d_in[5];
  const float* proj_b = (const float*)d_in[6];
  const float* temperature = (const float*)d_in[7];
  const float* sp_w = (const float*)d_in[8];
  const float* sp_b = (const float*)d_in[9];
  const float* up_w = (const float*)d_in[10];
  const float* up_b = (const float*)d_in[11];
  const float* c2_w = (const float*)d_in[12];
  const float* c2_b = (const float*)d_in[13];
  const float* ln2_w = (const float*)d_in[14];
  const float* ln2_b = (const float*)d_in[15];
  const float* c4_w = (const float*)d_in[16];
  const float* c4_b = (const float*)d_in[17];
  const float* ln4_w = (const float*)d_in[18];
  const float* ln4_b = (const float*)d_in[19];
  const float* c6_w = (const float*)d_in[20];
  const float* c6_b = (const float*)d_in[21];
  const float* ln6_w = (const float*)d_in[22];
  const float* ln6_b = (const float*)d_in[23];
  const float* mp_w = (const float*)d_in[24];
  const float* mp_b = (const float*)d_in[25];
  float* ws = (float*)d_ws;
  float* out = (float*)d_out;

  k1_xseq<<<dim3(WLEN / TILE, BATCH), 256, 0, stream>>>(x, qkv_w, qkv_b, dw_w,
                                                        dw_b, sp_w, sp_b, ws);
  k2_attn<<<dim3(HEADS, BATCH), 256, 0, stream>>>(
      x, qkv_w, qkv_b, dw_w, dw_b, temperature, mp_w, mp_b, ws);
  k3_mtl<<<dim3(BATCH), 256, 0, stream>>>(up_w, up_b, c2_w, c2_b, ln2_w, ln2_b,
                                          c4_w, c4_b, ln4_w, ln4_b, c6_w, c6_b,
                                          ln6_w, ln6_b, ws);
  k3b_combine<<<dim3(BATCH), 64, 0, stream>>>(proj_w, proj_b, ws);
  k4_out<<<dim3(WLEN / 1024, BATCH), 256, 0, stream>>>(ws, out);
}